// GatD2rlActor_64304250356439
// MI455X (gfx1250) — compile-verified
//
#include <hip/hip_runtime.h>
#include <hip/hip_bf16.h>

#define NN 100000
#define EE 3200000
#define FIN 128
#define CC 16
#define BB 512
#define OXY 64
#define ROTD 4

typedef float v2f __attribute__((ext_vector_type(2)));
typedef float v8f __attribute__((ext_vector_type(8)));

// ---------------- workspace layout (float offsets) ----------------
#define OFF_H    ((size_t)0)                    // N*16  h1 then h2
#define OFF_Y1   (OFF_H   + (size_t)NN*CC)      // N*16  relu(gat1) pre-BN
#define OFF_ACC  (OFF_Y1  + (size_t)NN*CC)      // N*16  softmax numerator acc
#define OFF_AS   (OFF_ACC + (size_t)NN*CC)      // N
#define OFF_AD   (OFF_AS  + (size_t)NN)         // N
#define OFF_M    (OFF_AD  + (size_t)NN)         // N (uint, encoded max)
#define OFF_SL   (OFF_M   + (size_t)NN)         // N self-loop logit
#define OFF_DEN  (OFF_SL  + (size_t)NN)         // N
#define OFF_DEG  (OFF_DEN + (size_t)NN)         // N
#define OFF_S1   (OFF_DEG + (size_t)NN)         // N  sum(edge_attr.v1) per dst
#define OFF_S2   (OFF_S1  + (size_t)NN)         // N
#define OFF_AE1  (OFF_S2  + (size_t)NN)         // E  per-edge a_e layer1
#define OFF_AE2  (OFF_AE1 + (size_t)EE)         // E
#define OFF_V    (OFF_AE2 + (size_t)EE)         // 4  (We1@ae1, We2@ae2)
#define OFF_BN   (OFF_V   + (size_t)4)          // 64: sum16, sq16, scale16, shift16
#define OFF_XE   (OFF_BN  + (size_t)64)         // B*16
#define OFF_CNT  (OFF_XE  + (size_t)BB*CC)      // B
#define OFF_AUX1 (OFF_CNT + (size_t)BB)         // 128*16 zero-padded [W1@as1|W1@ad1|0..]
#define OFF_AUX2 (OFF_AUX1 + (size_t)FIN*CC)    // 16*16  zero-padded [W2@as2|W2@ad2|0..]

__device__ __forceinline__ unsigned enc_f(float f){
  unsigned b = __float_as_uint(f);
  return (b & 0x80000000u) ? ~b : (b | 0x80000000u);
}
__device__ __forceinline__ float dec_f(unsigned k){
  return __uint_as_float((k & 0x80000000u) ? (k & 0x7fffffffu) : ~k);
}
__device__ __forceinline__ float lrelu(float x){ return x >= 0.f ? x : 0.2f * x; }

// precompute: aux1 = [W1@as1 | W1@ad1 | 0...] (128x16, zero-padded cols),
//             aux2 = [W2@as2 | W2@ad2 | 0...] (16x16),
//             v    = {We1@ae1 (2), We2@ae2 (2)}
__global__ void k_aux(const float* __restrict__ W1, const float* __restrict__ as1,
                      const float* __restrict__ ad1, const float* __restrict__ W2,
                      const float* __restrict__ as2, const float* __restrict__ ad2,
                      const float* __restrict__ We1, const float* __restrict__ ae1,
                      const float* __restrict__ We2, const float* __restrict__ ae2,
                      float* __restrict__ aux1, float* __restrict__ aux2,
                      float* __restrict__ v){
  int t = blockIdx.x * blockDim.x + threadIdx.x;
  if (t < FIN*CC){
    int k = t >> 4, c = t & 15;
    float a = 0.f;
    if (c < 2){
      const float* att = c ? ad1 : as1;
      for (int j = 0; j < CC; ++j) a += W1[k*CC + j] * att[j];
    }
    aux1[t] = a;
  } else if (t < FIN*CC + CC*CC){
    int u = t - FIN*CC;
    int k = u >> 4, c = u & 15;
    float a = 0.f;
    if (c < 2){
      const float* att = c ? ad2 : as2;
      for (int j = 0; j < CC; ++j) a += W2[k*CC + j] * att[j];
    }
    aux2[u] = a;
  } else if (t < FIN*CC + CC*CC + 4){
    int u = t - FIN*CC - CC*CC;
    const float* We = (u < 2) ? We1 : We2;
    const float* ae = (u < 2) ? ae1 : ae2;
    int k = u & 1;
    float a = 0.f;
    for (int j = 0; j < CC; ++j) a += We[k*CC + j] * ae[j];
    v[u] = a;
  }
}

// per-edge a_e for both layers + deg + self-loop attr sums (dot folded, linear)
__global__ void k_edge_prep(const int* __restrict__ ei, const float* __restrict__ ea,
                            const float* __restrict__ v, float* __restrict__ ae1e,
                            float* __restrict__ ae2e, float* __restrict__ s1,
                            float* __restrict__ s2, float* __restrict__ deg){
  int e = blockIdx.x * blockDim.x + threadIdx.x;
  if (e >= EE) return;
  int dst = ei[EE + e];
  float e0 = ea[2*e], e1 = ea[2*e + 1];
  float d1 = e0 * v[0] + e1 * v[1];
  float d2 = e0 * v[2] + e1 * v[3];
  ae1e[e] = d1; ae2e[e] = d2;
  atomicAdd(&s1[dst], d1);
  atomicAdd(&s2[dst], d2);
  atomicAdd(&deg[dst], 1.f);
}

// store D tile rows; a_s/a_d come from the aux-accumulator's cols 0/1
__device__ __forceinline__ void wmma_epilogue(v8f c, v8f cx, int tile, int half, int l16,
                                              float* __restrict__ h,
                                              float* __restrict__ a_s,
                                              float* __restrict__ a_d){
  #pragma unroll
  for (int j = 0; j < 8; ++j)
    h[(size_t)(tile*16 + j + half*8)*CC + l16] = c[j];
  if (l16 < 2){
    float* dst = l16 ? a_d : a_s;
    #pragma unroll
    for (int j = 0; j < 8; ++j) dst[tile*16 + j + half*8] = cx[j];
  }
}

// h1 = x @ W1 ([N,128]@[128,16]) via V_WMMA_F32_16X16X4_F32;
// a_s/a_d fused as second WMMA against zero-padded [W1@as1|W1@ad1] columns
__global__ void k_gemm1(const float* __restrict__ x, const float* __restrict__ W1,
                        const float* __restrict__ aux1, float* __restrict__ h,
                        float* __restrict__ a_s, float* __restrict__ a_d){
  int wave = (blockIdx.x * blockDim.x + threadIdx.x) >> 5;
  int lane = threadIdx.x & 31;
  if (wave >= NN/16) return;
  int half = lane >> 4, l16 = lane & 15;
  const float* xr = x + ((size_t)(wave*16 + l16)) * FIN;
  v8f c = {}, cx = {};
  #pragma unroll 4
  for (int ks = 0; ks < FIN/4; ++ks){
    int k0 = ks*4 + half*2;
    v2f a, b, ba;
    a.x  = xr[k0];                 a.y  = xr[k0 + 1];
    b.x  = W1[k0*CC + l16];        b.y  = W1[(k0+1)*CC + l16];
    ba.x = aux1[k0*CC + l16];      ba.y = aux1[(k0+1)*CC + l16];
    c  = __builtin_amdgcn_wmma_f32_16x16x4_f32(false, a, false, b,  (short)0, c,  false, false);
    cx = __builtin_amdgcn_wmma_f32_16x16x4_f32(false, a, false, ba, (short)0, cx, false, false);
  }
  wmma_epilogue(c, cx, wave, half, l16, h, a_s, a_d);
}

// h2 = BN(y1) @ W2 (BN folded to per-channel affine on A-load); fused a_s/a_d
__global__ void k_gemm2(const float* __restrict__ y1, const float* __restrict__ W2,
                        const float* __restrict__ aux2, const float* __restrict__ bn,
                        float* __restrict__ h, float* __restrict__ a_s,
                        float* __restrict__ a_d){
  int wave = (blockIdx.x * blockDim.x + threadIdx.x) >> 5;
  int lane = threadIdx.x & 31;
  if (wave >= NN/16) return;
  int half = lane >> 4, l16 = lane & 15;
  const float* yr = y1 + ((size_t)(wave*16 + l16)) * CC;
  const float* sc = bn + 32; const float* sh = bn + 48;
  v8f c = {}, cx = {};
  #pragma unroll
  for (int ks = 0; ks < 4; ++ks){
    int k0 = ks*4 + half*2;
    v2f a, b, ba;
    a.x  = yr[k0]   * sc[k0]   + sh[k0];
    a.y  = yr[k0+1] * sc[k0+1] + sh[k0+1];
    b.x  = W2[k0*CC + l16];    b.y  = W2[(k0+1)*CC + l16];
    ba.x = aux2[k0*CC + l16];  ba.y = aux2[(k0+1)*CC + l16];
    c  = __builtin_amdgcn_wmma_f32_16x16x4_f32(false, a, false, b,  (short)0, c,  false, false);
    cx = __builtin_amdgcn_wmma_f32_16x16x4_f32(false, a, false, ba, (short)0, cx, false, false);
  }
  wmma_epilogue(c, cx, wave, half, l16, h, a_s, a_d);
}

// seed segment-max with self-loop logit, stash it for finalize
__global__ void k_self(const float* __restrict__ s, const float* __restrict__ deg,
                       const float* __restrict__ a_s, const float* __restrict__ a_d,
                       unsigned* __restrict__ m, float* __restrict__ sl){
  int n = blockIdx.x * blockDim.x + threadIdx.x;
  if (n >= NN) return;
  float le = s[n] / fmaxf(deg[n], 1.f);
  float l = lrelu(a_s[n] + a_d[n] + le);
  sl[n] = l;
  m[n] = enc_f(l);
}

__global__ void k_edge_max(const int* __restrict__ ei, const float* __restrict__ aee,
                           const float* __restrict__ a_s, const float* __restrict__ a_d,
                           unsigned* __restrict__ m){
  int e = blockIdx.x * blockDim.x + threadIdx.x;
  if (e >= EE) return;
  int src = ei[e], dst = ei[EE + e];
  float l = lrelu(a_s[src] + a_d[dst] + aee[e]);
  atomicMax(&m[dst], enc_f(l));
}

// 16 lanes per edge: accumulate denom and w*h[src] (numerator); divide later
__global__ void k_edge_accum(const int* __restrict__ ei, const float* __restrict__ aee,
                             const float* __restrict__ a_s, const float* __restrict__ a_d,
                             const unsigned* __restrict__ m, const float* __restrict__ h,
                             float* __restrict__ acc, float* __restrict__ den){
  int t = blockIdx.x * blockDim.x + threadIdx.x;
  int e = t >> 4, c = t & 15;
  if (e >= EE) return;
  int src = ei[e], dst = ei[EE + e];
  float l = lrelu(a_s[src] + a_d[dst] + aee[e]);
  float w = __expf(l - dec_f(m[dst]));
  atomicAdd(&acc[(size_t)dst*CC + c], w * h[(size_t)src*CC + c]);
  if (c == 0) atomicAdd(&den[dst], w);
}

// finalize layer1: add self-loop, divide, +bias, relu, BN stats (block-reduced)
__global__ void k_fin1(const float* __restrict__ acc, const float* __restrict__ den,
                       const unsigned* __restrict__ m, const float* __restrict__ sl,
                       const float* __restrict__ h, const float* __restrict__ b1,
                       float* __restrict__ y1, float* __restrict__ bnstat){
  __shared__ float red[512];
  int t = threadIdx.x, c = t & 15, g = t >> 4;
  int base = blockIdx.x * 256;
  float bc = b1[c], s = 0.f, q = 0.f;
  for (int i = 0; i < 16; ++i){
    int n = base + i*16 + g;
    if (n < NN){
      float wsf = __expf(sl[n] - dec_f(m[n]));
      float d = den[n] + wsf;
      float o = (acc[(size_t)n*CC + c] + wsf * h[(size_t)n*CC + c]) / d + bc;
      o = fmaxf(o, 0.f);
      y1[(size_t)n*CC + c] = o;
      s += o; q += o * o;
    }
  }
  red[t] = s; red[256 + t] = q;
  __syncthreads();
  if (t < 16){
    float ss = 0.f, qq = 0.f;
    for (int gg = 0; gg < 16; ++gg){ ss += red[gg*16 + t]; qq += red[256 + gg*16 + t]; }
    atomicAdd(&bnstat[t], ss);
    atomicAdd(&bnstat[16 + t], qq);
  }
}

// BN stats -> per-channel affine
__global__ void k_bnaff(const float* __restrict__ g1, const float* __restrict__ bb1,
                        float* __restrict__ bn){
  int c = threadIdx.x;
  if (c >= CC) return;
  float mu = bn[c] / (float)NN;
  float var = bn[16 + c] / (float)NN - mu * mu;
  float sc = g1[c] * rsqrtf(var + 1e-5f);
  bn[32 + c] = sc;
  bn[48 + c] = bb1[c] - mu * sc;
}

// finalize layer2 and scatter straight into per-graph mean pool
__global__ void k_fin2(const float* __restrict__ acc, const float* __restrict__ den,
                       const unsigned* __restrict__ m, const float* __restrict__ sl,
                       const float* __restrict__ h, const float* __restrict__ b2,
                       const int* __restrict__ batch, float* __restrict__ xe,
                       float* __restrict__ cnt){
  int t = threadIdx.x, c = t & 15, g = t >> 4;
  int base = blockIdx.x * 256;
  float bc = b2[c];
  for (int i = 0; i < 16; ++i){
    int n = base + i*16 + g;
    if (n < NN){
      float wsf = __expf(sl[n] - dec_f(m[n]));
      float d = den[n] + wsf;
      float o = (acc[(size_t)n*CC + c] + wsf * h[(size_t)n*CC + c]) / d + bc;
      o = fmaxf(o, 0.f);
      int b = batch[n];
      atomicAdd(&xe[b*CC + c], o);
      if (c == 0) atomicAdd(&cnt[b], 1.f);
    }
  }
}

// entire D2RL tail: one block, one thread per graph
__global__ void __launch_bounds__(512)
k_tail(const float* __restrict__ xe, const float* __restrict__ cnt,
       const float* __restrict__ gL1, const float* __restrict__ bL1,
       const float* __restrict__ Wl1, const float* __restrict__ bl1,
       const float* __restrict__ gL2, const float* __restrict__ bL2,
       const float* __restrict__ Wl2, const float* __restrict__ bl2,
       const float* __restrict__ gL3, const float* __restrict__ bL3,
       const float* __restrict__ Wl3, const float* __restrict__ bl3,
       const float* __restrict__ Wx, const float* __restrict__ bx,
       const float* __restrict__ Wy, const float* __restrict__ by,
       const float* __restrict__ Wr, const float* __restrict__ br,
       float* __restrict__ out){
  __shared__ float ssum[32], ssq[32];
  int b = threadIdx.x;

  auto bn_block = [&](float* v, int nc, const float* g, const float* bb){
    if (b < 32){ ssum[b] = 0.f; ssq[b] = 0.f; }
    __syncthreads();
    for (int c = 0; c < nc; ++c){
      float xv = v[c], s = xv, q = xv * xv;
      #pragma unroll
      for (int o = 16; o >= 1; o >>= 1){ s += __shfl_xor(s, o); q += __shfl_xor(q, o); }
      if ((b & 31) == 0){ atomicAdd(&ssum[c], s); atomicAdd(&ssq[c], q); }
    }
    __syncthreads();
    for (int c = 0; c < nc; ++c){
      float mu = ssum[c] * (1.f / (float)BB);
      float var = ssq[c] * (1.f / (float)BB) - mu * mu;
      v[c] = g[c] * (v[c] - mu) * rsqrtf(var + 1e-5f) + bb[c];
    }
    __syncthreads();
  };

  float xr[CC];
  float inv = 1.f / fmaxf(cnt[b], 1.f);
  #pragma unroll
  for (int c = 0; c < CC; ++c) xr[c] = xe[b*CC + c] * inv;

  float z0[CC];
  #pragma unroll
  for (int c = 0; c < CC; ++c) z0[c] = xr[c];
  bn_block(z0, CC, gL1, bL1);

  float z1[CC];
  #pragma unroll
  for (int o = 0; o < CC; ++o){
    float a = bl1[o];
    for (int c = 0; c < CC; ++c) a += z0[c] * Wl1[c*CC + o];
    z1[o] = fmaxf(a, 0.f);
  }

  float v2[2*CC];
  #pragma unroll
  for (int c = 0; c < CC; ++c){ v2[c] = z1[c]; v2[CC + c] = xr[c]; }
  bn_block(v2, 2*CC, gL2, bL2);

  float z2[CC];
  #pragma unroll
  for (int o = 0; o < CC; ++o){
    float a = bl2[o];
    for (int c = 0; c < 2*CC; ++c) a += v2[c] * Wl2[c*CC + o];
    z2[o] = fmaxf(a, 0.f);
  }

  float v3[2*CC];
  #pragma unroll
  for (int c = 0; c < CC; ++c){ v3[c] = z2[c]; v3[CC + c] = xr[c]; }
  bn_block(v3, 2*CC, gL3, bL3);

  float z3[CC];
  #pragma unroll
  for (int o = 0; o < CC; ++o){
    float a = bl3[o];
    for (int c = 0; c < 2*CC; ++c) a += v3[c] * Wl3[c*CC + o];
    z3[o] = fmaxf(a, 0.f);
  }

  // streaming softmax heads (3 passes: max, sum, write) to keep VGPRs low
  auto head64 = [&](const float* W, const float* bs, float* o64){
    float mx = -3.0e38f;
    for (int o = 0; o < OXY; ++o){
      float a = bs[o];
      for (int c = 0; c < CC; ++c) a += z3[c] * W[c*OXY + o];
      mx = fmaxf(mx, a);
    }
    float sum = 0.f;
    for (int o = 0; o < OXY; ++o){
      float a = bs[o];
      for (int c = 0; c < CC; ++c) a += z3[c] * W[c*OXY + o];
      sum += __expf(a - mx);
    }
    float isum = 1.f / sum;
    for (int o = 0; o < OXY; ++o){
      float a = bs[o];
      for (int c = 0; c < CC; ++c) a += z3[c] * W[c*OXY + o];
      o64[o] = __expf(a - mx) * isum;
    }
  };
  head64(Wx, bx, out + (size_t)b*OXY);
  head64(Wy, by, out + (size_t)BB*OXY + (size_t)b*OXY);

  float lr_[ROTD];
  float mx = -3.0e38f;
  #pragma unroll
  for (int o = 0; o < ROTD; ++o){
    float a = br[o];
    for (int c = 0; c < CC; ++c) a += z3[c] * Wr[c*ROTD + o];
    lr_[o] = a; mx = fmaxf(mx, a);
  }
  float sum = 0.f;
  #pragma unroll
  for (int o = 0; o < ROTD; ++o){ lr_[o] = __expf(lr_[o] - mx); sum += lr_[o]; }
  float isum = 1.f / sum;
  float* ro = out + 2*(size_t)BB*OXY + (size_t)b*ROTD;
  #pragma unroll
  for (int o = 0; o < ROTD; ++o) ro[o] = lr_[o] * isum;
}

extern "C" void kernel_launch(void* const* d_in, const int* in_sizes, int n_in,
                              void* d_out, int out_size, void* d_ws, size_t ws_size,
                              hipStream_t stream) {
  const float* x   = (const float*)d_in[0];
  const int*   ei  = (const int*)  d_in[1];
  const float* ea  = (const float*)d_in[2];
  const int*   bat = (const int*)  d_in[3];
  const float* W1  = (const float*)d_in[4];
  const float* We1 = (const float*)d_in[5];
  const float* as1 = (const float*)d_in[6];
  const float* ad1 = (const float*)d_in[7];
  const float* ae1 = (const float*)d_in[8];
  const float* b1  = (const float*)d_in[9];
  const float* g1  = (const float*)d_in[10];
  const float* bb1 = (const float*)d_in[11];
  const float* W2  = (const float*)d_in[12];
  const float* We2 = (const float*)d_in[13];
  const float* as2 = (const float*)d_in[14];
  const float* ad2 = (const float*)d_in[15];
  const float* ae2 = (const float*)d_in[16];
  const float* b2  = (const float*)d_in[17];
  const float* gL1 = (const float*)d_in[18];
  const float* bL1 = (const float*)d_in[19];
  const float* Wl1 = (const float*)d_in[20];
  const float* bl1 = (const float*)d_in[21];
  const float* gL2 = (const float*)d_in[22];
  const float* bL2 = (const float*)d_in[23];
  const float* Wl2 = (const float*)d_in[24];
  const float* bl2 = (const float*)d_in[25];
  const float* gL3 = (const float*)d_in[26];
  const float* bL3 = (const float*)d_in[27];
  const float* Wl3 = (const float*)d_in[28];
  const float* bl3 = (const float*)d_in[29];
  const float* Wx  = (const float*)d_in[30];
  const float* bx  = (const float*)d_in[31];
  const float* Wy  = (const float*)d_in[32];
  const float* by  = (const float*)d_in[33];
  const float* Wr  = (const float*)d_in[34];
  const float* br  = (const float*)d_in[35];

  float* ws = (float*)d_ws;
  float*    h    = ws + OFF_H;
  float*    y1   = ws + OFF_Y1;
  float*    acc  = ws + OFF_ACC;
  float*    a_s  = ws + OFF_AS;
  float*    a_d  = ws + OFF_AD;
  unsigned* m    = (unsigned*)(ws + OFF_M);
  float*    sl   = ws + OFF_SL;
  float*    den  = ws + OFF_DEN;
  float*    deg  = ws + OFF_DEG;
  float*    s1   = ws + OFF_S1;
  float*    s2   = ws + OFF_S2;
  float*    ae1e = ws + OFF_AE1;
  float*    ae2e = ws + OFF_AE2;
  float*    v    = ws + OFF_V;
  float*    bn   = ws + OFF_BN;
  float*    xe   = ws + OFF_XE;
  float*    cnt  = ws + OFF_CNT;
  float*    aux1 = ws + OFF_AUX1;
  float*    aux2 = ws + OFF_AUX2;

  const int EB = (EE + 255) / 256;                    // edge-parallel blocks
  const int NB = (NN + 255) / 256;                    // node-parallel blocks
  const int GB = (NN/16 + 7) / 8;                     // GEMM: 8 tiles per 256-thr block
  const int AB = (int)(((size_t)EE*16 + 255) / 256);  // 16 lanes per edge
  const int XB = (FIN*CC + CC*CC + 4 + 255) / 256;    // precompute threads

  // zero: acc .. s2 (contiguous), and v/bn/xe/cnt (contiguous)
  hipMemsetAsync(acc, 0, (size_t)(OFF_S2 + NN - OFF_ACC) * sizeof(float), stream);
  hipMemsetAsync(v, 0, (size_t)(4 + 64 + BB*CC + BB) * sizeof(float), stream);

  k_aux      <<<XB, 256, 0, stream>>>(W1, as1, ad1, W2, as2, ad2,
                                      We1, ae1, We2, ae2, aux1, aux2, v);
  k_edge_prep<<<EB, 256, 0, stream>>>(ei, ea, v, ae1e, ae2e, s1, s2, deg);

  // ---- GAT layer 1 ----
  k_gemm1    <<<GB, 256, 0, stream>>>(x, W1, aux1, h, a_s, a_d);
  k_self     <<<NB, 256, 0, stream>>>(s1, deg, a_s, a_d, m, sl);
  k_edge_max <<<EB, 256, 0, stream>>>(ei, ae1e, a_s, a_d, m);
  k_edge_accum<<<AB, 256, 0, stream>>>(ei, ae1e, a_s, a_d, m, h, acc, den);
  k_fin1     <<<NB, 256, 0, stream>>>(acc, den, m, sl, h, b1, y1, bn);
  k_bnaff    <<<1, 32, 0, stream>>>(g1, bb1, bn);

  // ---- GAT layer 2 (reuse acc/den/m/sl/a_s/a_d/h) ----
  hipMemsetAsync(acc, 0, (size_t)NN * CC * sizeof(float), stream);
  hipMemsetAsync(den, 0, (size_t)NN * sizeof(float), stream);
  k_gemm2    <<<GB, 256, 0, stream>>>(y1, W2, aux2, bn, h, a_s, a_d);
  k_self     <<<NB, 256, 0, stream>>>(s2, deg, a_s, a_d, m, sl);
  k_edge_max <<<EB, 256, 0, stream>>>(ei, ae2e, a_s, a_d, m);
  k_edge_accum<<<AB, 256, 0, stream>>>(ei, ae2e, a_s, a_d, m, h, acc, den);
  k_fin2     <<<NB, 256, 0, stream>>>(acc, den, m, sl, h, b2, bat, xe, cnt);

  // ---- pooled mean + D2RL tail + softmax heads ----
  k_tail<<<1, 512, 0, stream>>>(xe, cnt, gL1, bL1, Wl1, bl1, gL2, bL2, Wl2, bl2,
                                gL3, bL3, Wl3, bl3, Wx, bx, Wy, by, Wr, br,
                                (float*)d_out);
}